// GCNEncoder_24455543783860
// MI455X (gfx1250) — compile-verified
//
#include <hip/hip_runtime.h>
#include <hip/hip_bf16.h>

// ---------------------------------------------------------------------------
// 2-layer GCN for MI455X (gfx1250, wave32).
//  - Dense transforms: V_WMMA_F32_16X16X4_F32, weights staged to LDS via the
//    Tensor Data Mover (tensor_load_to_lds + s_wait_tensorcnt), LDS rows
//    padded to 80 dwords so B-operand ds_load_2addr hits disjoint banks.
//  - Edge aggregation: float4 gathers (global_load_b128) + fire-and-forget
//    global_atomic_add_f32 resolved in L2 (all features fit in 192MB L2).
//  - ReLU is a template parameter: no per-element cndmask chains.
// ---------------------------------------------------------------------------

#define D 64          // feature dim
#define LDSTRIDE 80   // padded LDS row stride in floats (80 = 16 mod 64)

typedef __attribute__((ext_vector_type(2))) float v2f;
typedef __attribute__((ext_vector_type(8))) float v8f;
typedef __attribute__((ext_vector_type(4))) unsigned int u32x4;
typedef __attribute__((ext_vector_type(4))) int i32x4;
typedef __attribute__((ext_vector_type(8))) int i32x8;

// ---- degree / normalization ------------------------------------------------

__global__ void deg_init_kernel(float* __restrict__ deg, int n) {
  int i = blockIdx.x * blockDim.x + threadIdx.x;
  if (i < n) deg[i] = 1.0f;  // self-loop contributes 1 to in-degree
}

__global__ void deg_accum_kernel(const long long* __restrict__ dst,
                                 float* __restrict__ deg, int e) {
  int i = blockIdx.x * blockDim.x + threadIdx.x;
  if (i < e) unsafeAtomicAdd(&deg[(int)dst[i]], 1.0f);
}

__global__ void dinv_kernel(float* __restrict__ deg, int n) {
  int i = blockIdx.x * blockDim.x + threadIdx.x;
  if (i < n) deg[i] = rsqrtf(deg[i]);  // deg >= 1 always (self-loop)
}

// ---- TDM staging of W (64x64 f32) into LDS with row padding ----------------
// D# per CDNA5 ISA:
//   group0: count=1 | lds_addr | global_addr[31:0] | global_addr[56:32]+type=2
//   group1: data_size=4B, pad_enable, pad_interval=5 (64 dwords),
//           pad_amount=15 (16 dwords) -> LDS row stride 80 dwords;
//           tensor_dim0=64, tensor_dim1=64, tile=64x64, dim0_stride=64.

__device__ __forceinline__ void stage_W_lds(float* lds, const float* W) {
#if defined(__has_builtin) && __has_builtin(__builtin_amdgcn_tensor_load_to_lds)
  if ((threadIdx.x >> 5) == 0) {  // one wave issues the DMA (EXEC ignored)
    unsigned lds_addr = (unsigned)(unsigned long long)(uintptr_t)lds;
    unsigned long long ga = (unsigned long long)(uintptr_t)W;
    u32x4 g0 = { 1u,                                   // count=1, user mode
                 lds_addr,
                 (unsigned)ga,
                 (unsigned)((ga >> 32) & 0x1FFFFFFull) | (2u << 30) };  // type=2
    i32x8 g1 = { (int)((2u << 16) | (1u << 20) | (5u << 22) | (15u << 25)),
                 (int)(64u << 16),   // tensor_dim0[15:0] @ bits 63:48
                 (int)(64u << 16),   // tensor_dim1[15:0] @ bits 95:80
                 (int)(64u << 16),   // tile_dim0        @ bits 127:112
                 64,                 // tile_dim1        @ bits 143:128
                 64,                 // tensor_dim0_stride[31:0]
                 0, 0 };
    i32x4 z4 = { 0, 0, 0, 0 };
#if __clang_major__ >= 23
    i32x8 z8 = { 0, 0, 0, 0, 0, 0, 0, 0 };
    __builtin_amdgcn_tensor_load_to_lds(g0, g1, z4, z4, z8, 0);
#else
    __builtin_amdgcn_tensor_load_to_lds(g0, g1, z4, z4, 0);
#endif
    __builtin_amdgcn_s_wait_tensorcnt(0);
  }
#else
  for (int i = threadIdx.x; i < D * D; i += blockDim.x) {
    int r = i >> 6, c = i & 63;
    lds[r * LDSTRIDE + c] = W[i];
  }
#endif
  __syncthreads();
}

// ---- fused GEMM: H = act(X) @ W ; OUT = H * dinv^2 + bias ------------------
// One wave computes a 16-row stripe (4 x 16x16 tiles, K=64).
// f32 WMMA VGPR layouts per CDNA5 ISA 7.12.2:
//   A 16x4 : lanes 0-15 row M=l, VGPRs = K{k,k+1}; lanes 16-31 K{k+2,k+3}
//   B 4x16 : lanes 0-15 col N=l, VGPRs = K{k,k+1}; lanes 16-31 K{k+2,k+3}
//   C 16x16: VGPR v -> row (v | v+8), col = lane&15

template <bool RELU_IN>
__global__ void gemm_wmma_fused_kernel(const float* __restrict__ X,
                                       const float* __restrict__ W,
                                       const float* __restrict__ bias,
                                       const float* __restrict__ dinv,
                                       float* __restrict__ H,
                                       float* __restrict__ OUT,
                                       int n) {
  __shared__ float Wlds[D * LDSTRIDE];  // 20 KB of 320 KB/WGP
  stage_W_lds(Wlds, W);

  const int wave = blockIdx.x * (blockDim.x >> 5) + (threadIdx.x >> 5);
  const int lane = threadIdx.x & 31;
  const int row0 = wave * 16;
  if (row0 >= n) return;

  const int half = lane >> 4;      // 0: lanes 0-15, 1: lanes 16-31
  const int l16  = lane & 15;
  int arow = row0 + l16;
  if (arow >= n) arow = n - 1;     // clamp (dup rows land in unstored D rows)
  const float* __restrict__ xrow = X + (size_t)arow * D;

  // preload + (optionally) relu the 64-float A row slice for this lane
  float areg[D / 2];
#pragma unroll
  for (int k = 0; k < D; k += 4) {
    float ax = xrow[k + half * 2];
    float ay = xrow[k + half * 2 + 1];
    if (RELU_IN) { ax = fmaxf(ax, 0.0f); ay = fmaxf(ay, 0.0f); }
    areg[k / 2]     = ax;
    areg[k / 2 + 1] = ay;
  }
  // preload dinv for the 8 output rows this lane stores
  float dloc[8];
#pragma unroll
  for (int v = 0; v < 8; ++v) {
    int m = row0 + v + half * 8;
    dloc[v] = dinv[m < n ? m : n - 1];
  }

  for (int nt = 0; nt < D / 16; ++nt) {
    const int col = nt * 16 + l16;
    v8f acc = {};
#pragma unroll
    for (int k = 0; k < D; k += 4) {
      const int ka = k + half * 2;
      v2f a, b;
      a.x = areg[k / 2];
      a.y = areg[k / 2 + 1];
      b.x = Wlds[ka * LDSTRIDE + col];
      b.y = Wlds[(ka + 1) * LDSTRIDE + col];
      acc = __builtin_amdgcn_wmma_f32_16x16x4_f32(
          false, a, false, b, (short)0, acc, false, false);
    }
    const float bcol = bias[col];
#pragma unroll
    for (int v = 0; v < 8; ++v) {
      const int m = row0 + v + half * 8;
      if (m < n) {
        const float hv = acc[v];
        const float di = dloc[v];
        H[(size_t)m * D + col]   = hv;                  // raw h (edge gather)
        OUT[(size_t)m * D + col] = hv * di * di + bcol; // self-loop + bias
      }
    }
  }
}

// ---- edge aggregation: OUT[dst] += H[src] * dinv[src]*dinv[dst] ------------
// Two edges per wave: 16 lanes x float4 (global_load_b128) per edge.
// Indices/norm loaded once per edge (lanes 0/16) and shfl-broadcast.

__global__ void edge_agg_kernel(const long long* __restrict__ src,
                                const long long* __restrict__ dst,
                                const float* __restrict__ dinv,
                                const float* __restrict__ H,
                                float* __restrict__ OUT, int e) {
  const int wid  = blockIdx.x * (blockDim.x >> 5) + (threadIdx.x >> 5);
  const int lane = threadIdx.x & 31;
  const int l16  = lane & 15;
  const int eidx = wid * 2 + (lane >> 4);
  if (eidx >= e) return;

  int s = 0, d = 0;
  float nr = 0.0f;
  if (l16 == 0) {
    s = (int)src[eidx];
    d = (int)dst[eidx];
    nr = dinv[s] * dinv[d];
  }
  const int leader = lane & 16;
  s  = __shfl(s,  leader);
  d  = __shfl(d,  leader);
  nr = __shfl(nr, leader);

  const float4 hv = ((const float4*)(H + (size_t)s * D))[l16];
  float* op = OUT + (size_t)d * D + l16 * 4;
  unsafeAtomicAdd(op + 0, hv.x * nr);
  unsafeAtomicAdd(op + 1, hv.y * nr);
  unsafeAtomicAdd(op + 2, hv.z * nr);
  unsafeAtomicAdd(op + 3, hv.w * nr);
}

// ---------------------------------------------------------------------------

extern "C" void kernel_launch(void* const* d_in, const int* in_sizes, int n_in,
                              void* d_out, int out_size, void* d_ws, size_t ws_size,
                              hipStream_t stream) {
  const float*     x   = (const float*)d_in[0];
  const long long* ei  = (const long long*)d_in[1];  // int64 [2, E]
  const float*     W1  = (const float*)d_in[2];
  const float*     b1  = (const float*)d_in[3];
  const float*     W2  = (const float*)d_in[4];
  const float*     b2  = (const float*)d_in[5];
  float*           out = (float*)d_out;

  const int n = in_sizes[0] / D;   // 100000
  const int e = in_sizes[1] / 2;   // 3200000
  const long long* src = ei;
  const long long* dst = ei + e;

  // workspace: deg/dinv | H1 | OUT1 | H2   (~77 MB)
  char* ws = (char*)d_ws;
  size_t off = 0;
  float* deg  = (float*)(ws + off); off += (((size_t)n * 4) + 255) & ~(size_t)255;
  float* H1   = (float*)(ws + off); off += (size_t)n * D * 4;
  float* OUT1 = (float*)(ws + off); off += (size_t)n * D * 4;
  float* H2   = (float*)(ws + off);

  const int TPB = 256;               // 8 waves / block
  const int WPB = TPB / 32;
  const int tiles  = (n + 15) / 16;  // 16-row stripes per GEMM wave
  const int ewaves = (e + 1) / 2;    // 2 edges per wave

  // normalization: deg -> dinv (in place)
  deg_init_kernel <<<(n + TPB - 1) / TPB, TPB, 0, stream>>>(deg, n);
  deg_accum_kernel<<<(e + TPB - 1) / TPB, TPB, 0, stream>>>(dst, deg, e);
  dinv_kernel     <<<(n + TPB - 1) / TPB, TPB, 0, stream>>>(deg, n);

  // layer 1: H1 = x@W1 ; OUT1 = H1*dinv^2 + b1 ; OUT1 += scatter(H1)
  gemm_wmma_fused_kernel<false><<<(tiles + WPB - 1) / WPB, TPB, 0, stream>>>(
      x, W1, b1, deg, H1, OUT1, n);
  edge_agg_kernel<<<(ewaves + WPB - 1) / WPB, TPB, 0, stream>>>(
      src, dst, deg, H1, OUT1, e);

  // layer 2: H2 = relu(OUT1)@W2 ; out = H2*dinv^2 + b2 ; out += scatter(H2)
  gemm_wmma_fused_kernel<true><<<(tiles + WPB - 1) / WPB, TPB, 0, stream>>>(
      OUT1, W2, b2, deg, H2, out, n);
  edge_agg_kernel<<<(ewaves + WPB - 1) / WPB, TPB, 0, stream>>>(
      src, dst, deg, H2, out, e);
}